// PolyphonicWaveletFilter_47648367182305
// MI455X (gfx1250) — compile-verified
//
#include <hip/hip_runtime.h>
#include <hip/hip_bf16.h>
#include <math.h>

// ---------------------------------------------------------------------------
// PolyphonicWaveletFilter on MI455X (gfx1250, wave32)
//
// Per head h:  S = X_h @ Wp_h   (GEMM via V_WMMA_F32_16X16X4_F32)
//   X_h  : (B*L, 256)  fp32 slice of x
//   Wp_h : (256, Npad) cols [0,K)=real taps, [K,2K)=imag taps, rest 0
// then a shifted tap-sum (the causal dilated conv) + atan2/sqrt epilogue.
// Scratch: S for ONE head at a time (B*L*Npad*4 <= 5.25 MB), head loop is
// stream-serialized so the same ws region is reused.
// ---------------------------------------------------------------------------

typedef __attribute__((ext_vector_type(2))) float v2f;
typedef __attribute__((ext_vector_type(8))) float v8f;

#define N_HEADS  16
#define HIDDEN   4096
#define HEAD_DIM 256
#define B_SZ     4
#define L_SZ     4096
#define MROWS    (B_SZ * L_SZ)   // 16384 GEMM rows
#define WAVES_PER_BLOCK 8

// ---------------------------------------------------------------------------
// Phase 1: S[b*L + l, col] = sum_c x[b,l,h,c] * Wp[c,col]
// One wave -> one 16-row M strip, all NT 16-col N tiles, K=256 in steps of 4.
// ---------------------------------------------------------------------------
template <int NT>
__global__ __launch_bounds__(256)
void dwconv_gemm_wmma(const float* __restrict__ x,   // (B*L, HIDDEN)
                      const float* __restrict__ wr,  // (HEAD_DIM, K)
                      const float* __restrict__ wi,  // (HEAD_DIM, K)
                      float* __restrict__ S,         // (B*L, NPAD)
                      int head, int K)
{
    constexpr int NPAD = NT * 16;
    extern __shared__ float ldsW[];                  // Wp: [256][NPAD]

    const int tid = threadIdx.x;

    // Cooperative pack of real|imag|zero-pad weights into LDS.
    for (int idx = tid; idx < HEAD_DIM * NPAD; idx += 256) {
        const int c   = idx / NPAD;
        const int col = idx - c * NPAD;
        float v = 0.0f;
        if (col < K)          v = wr[c * K + col];
        else if (col < 2 * K) v = wi[c * K + (col - K)];
        ldsW[idx] = v;
    }
    __syncthreads();

    const int lane  = tid & 31;
    const int wave  = tid >> 5;
    const int strip = blockIdx.x * WAVES_PER_BLOCK + wave;  // 16-row strip
    const int m     = lane & 15;
    const int kpair = (lane >> 4) << 1;                     // 0 or 2

    // A fragment source row (16x4 fp32 A layout: lanes 0-15 = K{0,1},
    // lanes 16-31 = K{2,3}, row M = lane&15).
    const float* xrow =
        x + (size_t)(strip * 16 + m) * HIDDEN + head * HEAD_DIM;

    v8f acc[NT];
    const v8f vzero = {};
#pragma unroll
    for (int t = 0; t < NT; ++t) acc[t] = vzero;

    for (int kb = 0; kb < HEAD_DIM; kb += 4) {
        const v2f a = *(const v2f*)(xrow + kb + kpair);
#pragma unroll
        for (int t = 0; t < NT; ++t) {
            const int coln = t * 16 + m;                    // B col = lane&15
            v2f bfrag;
            bfrag.x = ldsW[(kb + kpair)     * NPAD + coln]; // B row kb+kpair
            bfrag.y = ldsW[(kb + kpair + 1) * NPAD + coln]; // B row kb+kpair+1
            // 8 args: (neg_a, A, neg_b, B, c_mod, C, reuse_a, reuse_b)
            acc[t] = __builtin_amdgcn_wmma_f32_16x16x4_f32(
                false, a, false, bfrag, (short)0, acc[t], false, false);
        }
    }

    // D layout: VGPR i -> rows {i, i+8} for lane halves, col = lane&15.
    const int rowbase = strip * 16 + ((lane >> 4) << 3);
#pragma unroll
    for (int t = 0; t < NT; ++t) {
#pragma unroll
        for (int i = 0; i < 8; ++i) {
            S[(size_t)(rowbase + i) * NPAD + t * 16 + m] = acc[t][i];
        }
    }
}

// ---------------------------------------------------------------------------
// Phase 2: causal dilated tap-sum + mean + polar epilogue for one head.
//   U[l] = (1/256) * sum_k S[l-(K-1-k)*d, k]        (real cols)
//   V[l] = (1/256) * sum_k S[l-(K-1-k)*d, K+k]      (imag cols)
//   phi  = atan2(V, U+eps)   mag = sqrt(U^2+V^2+eps)
// Output: phi (B,16,L) then mag (B,16,L), concatenated flat.
// ---------------------------------------------------------------------------
__global__ __launch_bounds__(256)
void combine_polar(const float* __restrict__ S, float* __restrict__ out,
                   int head, int K, int dil, int NPAD)
{
    const int gid = blockIdx.x * 256 + threadIdx.x;   // b*L + l
    const int l   = gid & (L_SZ - 1);
    const int b   = gid >> 12;

    float u = 0.0f, v = 0.0f;
    for (int k = 0; k < K; ++k) {
        const int off = (K - 1 - k) * dil;
        if (l >= off) {
            const float* srow = S + (size_t)(gid - off) * NPAD;
            u += srow[k];
            v += srow[K + k];
        }
    }
    u *= (1.0f / 256.0f);
    v *= (1.0f / 256.0f);

    const float mag = sqrtf(u * u + v * v + 1e-8f);
    const float phi = atan2f(v, u + 1e-8f);

    const size_t o = (size_t)(b * N_HEADS + head) * L_SZ + l;
    out[o] = phi;
    out[(size_t)B_SZ * N_HEADS * L_SZ + o] = mag;
}

// ---------------------------------------------------------------------------
static const int KS[N_HEADS]  = {3,5,7,9,11,13,15,17,19,21,23,25,27,29,31,33};
static const int DIL[N_HEADS] = {1,1,2,2,4,4,8,8,1,2,4,8,16,16,32,32};

extern "C" void kernel_launch(void* const* d_in, const int* in_sizes, int n_in,
                              void* d_out, int out_size, void* d_ws, size_t ws_size,
                              hipStream_t stream)
{
    (void)in_sizes; (void)n_in; (void)out_size; (void)ws_size;

    // setup_inputs() order: x, real_weights[0..15], imag_weights[0..15]
    const float* x  = (const float*)d_in[0];
    float* S        = (float*)d_ws;      // per-head S, reused (<= 5.25 MB)
    float* out      = (float*)d_out;

    const dim3 blk(256);
    const dim3 grid1(MROWS / 16 / WAVES_PER_BLOCK);   // 128 blocks, 8 waves ea.
    const dim3 grid2(MROWS / 256);                    // 64 blocks

    for (int h = 0; h < N_HEADS; ++h) {
        const int K    = KS[h];
        const int d    = DIL[h];
        const int NT   = (2 * K + 15) / 16;
        const int NPAD = NT * 16;
        const float* wr = (const float*)d_in[1 + h];
        const float* wi = (const float*)d_in[1 + N_HEADS + h];
        const size_t shmem = (size_t)HEAD_DIM * NPAD * sizeof(float);

        switch (NT) {
        case 1: dwconv_gemm_wmma<1><<<grid1, blk, shmem, stream>>>(x, wr, wi, S, h, K); break;
        case 2: dwconv_gemm_wmma<2><<<grid1, blk, shmem, stream>>>(x, wr, wi, S, h, K); break;
        case 3: dwconv_gemm_wmma<3><<<grid1, blk, shmem, stream>>>(x, wr, wi, S, h, K); break;
        case 4: dwconv_gemm_wmma<4><<<grid1, blk, shmem, stream>>>(x, wr, wi, S, h, K); break;
        default: dwconv_gemm_wmma<5><<<grid1, blk, shmem, stream>>>(x, wr, wi, S, h, K); break;
        }
        // Stream ordering serializes phase1(h) -> phase2(h) -> phase1(h+1),
        // so the single S scratch region is safe to reuse.
        combine_polar<<<grid2, blk, 0, stream>>>(S, out, h, K, d, NPAD);
    }
}